// GATNetwork_77309411328711
// MI455X (gfx1250) — compile-verified
//
#include <hip/hip_runtime.h>
#include <hip/hip_bf16.h>

// ---------------------------------------------------------------------------
// GAT 2-layer network for MI455X (gfx1250, wave32).
// Dense projections: V_WMMA_F32_16X16X4_F32, one wave per 16-row strip with
// C/16 accumulators (A fragments loaded once, B fragments from LDS-staged W).
// Edge softmax/aggregation: L2 float atomics (global_atomic_add_f32,
// uint-key atomicMax for segment max), float4-vectorized gathers.
// ---------------------------------------------------------------------------

typedef float v2f __attribute__((ext_vector_type(2)));
typedef float v8f __attribute__((ext_vector_type(8)));

#define NEG_ATT 0.2f
#define NEG_ACT 0.01f
#define BN_EPS  1e-5f

// ---------------- WMMA GEMM: Y[M,16*CT] = X[M,K] @ W[K,16*CT] --------------
// One wave computes a 16 x (16*CT) strip. K stepped by 4 with
// V_WMMA_F32_16X16X4_F32; each A fragment (v2f) is reused across CT column
// tiles. W (K*C <= 8192 floats) staged in LDS by the whole block.
template <int CT>
__global__ void gemm_wmma_strip(const float* __restrict__ X,
                                const float* __restrict__ W,
                                float* __restrict__ Y,
                                int M, int K) {
    constexpr int C = CT * 16;
    __shared__ float Wl[8192];
    const int t  = threadIdx.x;
    const int kc = K * C;
    for (int i = t; i < kc; i += blockDim.x) Wl[i] = W[i];
    __syncthreads();

    const int wave = (blockIdx.x * blockDim.x + t) >> 5;
    const int lane = t & 31;
    if (wave >= (M >> 4)) return;            // wave-uniform: EXEC all-1 below
    const int row0 = wave << 4;

    // A 16x4 f32 layout: lanes 0-15 -> M=lane, K=k+0,k+1 ; lanes 16-31 -> K=k+2,k+3
    const int m  = lane & 15;
    const int kh = (lane >> 4) << 1;         // 0 or 2
    const float* xr = X + (size_t)(row0 + m) * K;

    v8f acc[CT];
#pragma unroll
    for (int i = 0; i < CT; ++i) acc[i] = (v8f){};

    for (int k = 0; k < K; k += 4) {
        if ((k & 31) == 0)                   // stream-ahead hint -> global_prefetch_b8
            __builtin_prefetch(xr + k + 64, 0, 0);
        v2f a = *(const v2f*)(xr + k + kh);  // one b64 load per K-step
#pragma unroll
        for (int ct = 0; ct < CT; ++ct) {
            v2f b;
            b.x = Wl[(k + kh)     * C + (ct << 4) + m];
            b.y = Wl[(k + kh + 1) * C + (ct << 4) + m];
            acc[ct] = __builtin_amdgcn_wmma_f32_16x16x4_f32(
                          false, a, false, b, (short)0, acc[ct], false, false);
        }
    }

    // C/D layout: acc[ct][r] at lane L -> row r + (L>=16 ? 8 : 0), col L&15
    const int rb = (lane >> 4) << 3;
#pragma unroll
    for (int ct = 0; ct < CT; ++ct) {
        float* yr = Y + (size_t)(row0 + rb) * C + (ct << 4) + m;
#pragma unroll
        for (int r = 0; r < 8; ++r) yr[(size_t)r * C] = acc[ct][r];
    }
}

// ---------------- per-node attention scores -------------------------------
__global__ void scores_kernel(const float* __restrict__ hx,
                              const float* __restrict__ a_src,
                              const float* __restrict__ a_dst,
                              float* __restrict__ as_, float* __restrict__ ad_,
                              int N, int H, int D) {
    int idx = blockIdx.x * blockDim.x + threadIdx.x;
    if (idx >= N * H) return;
    int n = idx / H, h = idx % H;
    const float* hp = hx + (size_t)n * H * D + (size_t)h * D;
    float ss = 0.f, sd = 0.f;
    for (int d = 0; d < D; ++d) {
        float v = hp[d];
        ss += v * a_src[h * D + d];
        sd += v * a_dst[h * D + d];
    }
    as_[idx] = ss;
    ad_[idx] = sd;
}

// ---------------- fills ----------------------------------------------------
__global__ void fill_f32(float* p, float v, long long n) {
    long long i = (long long)blockIdx.x * blockDim.x + threadIdx.x;
    if (i < n) p[i] = v;
}
__global__ void fill_u32(unsigned* p, unsigned v, long long n) {
    long long i = (long long)blockIdx.x * blockDim.x + threadIdx.x;
    if (i < n) p[i] = v;
}

// ---------------- edge helpers --------------------------------------------
__device__ __forceinline__ void edge_pair(const int* __restrict__ srcA,
                                          const int* __restrict__ dstA,
                                          int E, long long e, int& s, int& d) {
    if (e < E) { s = srcA[e]; d = dstA[e]; }
    else       { s = d = (int)(e - E); }      // virtual self loop
}
__device__ __forceinline__ unsigned f32_key(float f) {
    unsigned u = __float_as_uint(f);
    return u ^ (unsigned)(((int)u >> 31) | 0x80000000);
}
__device__ __forceinline__ float key_f32(unsigned k) {
    unsigned u = (k & 0x80000000u) ? (k ^ 0x80000000u) : ~k;
    return __uint_as_float(u);
}

// pass 1: segment max of leaky-relu edge scores (order-preserving uint keys)
__global__ void edge_max_kernel(const int* __restrict__ srcA, const int* __restrict__ dstA,
                                int E, int N,
                                const float* __restrict__ as_, const float* __restrict__ ad_,
                                unsigned* __restrict__ mkey, int H) {
    long long idx = (long long)blockIdx.x * blockDim.x + threadIdx.x;
    long long tot = (long long)(E + N) * H;
    if (idx >= tot) return;
    long long e = idx / H; int h = (int)(idx % H);
    int s, d; edge_pair(srcA, dstA, E, e, s, d);
    float sc = as_[(size_t)s * H + h] + ad_[(size_t)d * H + h];
    sc = sc > 0.f ? sc : sc * NEG_ATT;
    atomicMax(&mkey[(size_t)d * H + h], f32_key(sc));
}

// pass 2: alpha = exp(score - max); segment sum via float atomic add
__global__ void edge_sum_kernel(const int* __restrict__ srcA, const int* __restrict__ dstA,
                                int E, int N,
                                const float* __restrict__ as_, const float* __restrict__ ad_,
                                const unsigned* __restrict__ mkey,
                                float* __restrict__ alpha, float* __restrict__ ssum, int H) {
    long long idx = (long long)blockIdx.x * blockDim.x + threadIdx.x;
    long long tot = (long long)(E + N) * H;
    if (idx >= tot) return;
    long long e = idx / H; int h = (int)(idx % H);
    int s, d; edge_pair(srcA, dstA, E, e, s, d);
    float sc = as_[(size_t)s * H + h] + ad_[(size_t)d * H + h];
    sc = sc > 0.f ? sc : sc * NEG_ATT;
    float ex = __expf(sc - key_f32(mkey[(size_t)d * H + h]));
    alpha[idx] = ex;
    atomicAdd(&ssum[(size_t)d * H + h], ex);
}

// pass 3: out[dst,c..c+3] += hx[src,c..c+3] * alpha/(sum+eps)
// one thread per (edge, 4-channel group): b128 gather + 4 float atomic adds
__global__ void edge_agg_kernel(const int* __restrict__ srcA, const int* __restrict__ dstA,
                                int E, int N,
                                const float* __restrict__ hx,
                                const float* __restrict__ alpha,
                                const float* __restrict__ ssum,
                                float* __restrict__ out, int C, int D, int H) {
    long long idx = (long long)blockIdx.x * blockDim.x + threadIdx.x;
    const int cq = C >> 2;
    long long tot = (long long)(E + N) * cq;
    if (idx >= tot) return;
    long long e = idx / cq;
    int c = (int)(idx % cq) << 2;            // 4 divides D -> one head per group
    int h = c / D;
    int s, d; edge_pair(srcA, dstA, E, e, s, d);
    float a = alpha[e * H + h] / (ssum[(size_t)d * H + h] + 1e-16f);
    const float4 hv = *(const float4*)(hx + (size_t)s * C + c);
    float* op = out + (size_t)d * C + c;
    atomicAdd(op + 0, hv.x * a);
    atomicAdd(op + 1, hv.y * a);
    atomicAdd(op + 2, hv.z * a);
    atomicAdd(op + 3, hv.w * a);
}

// ---------------- bias + leaky_relu(0.01) + batchnorm (in place) ----------
__global__ void post_kernel(float* __restrict__ out, const float* __restrict__ bias,
                            const float* __restrict__ g, const float* __restrict__ be,
                            const float* __restrict__ mean, const float* __restrict__ var,
                            int N, int C) {
    long long idx = (long long)blockIdx.x * blockDim.x + threadIdx.x;
    if (idx >= (long long)N * C) return;
    int c = (int)(idx % C);
    float v = out[idx] + bias[c];
    v = v > 0.f ? v : v * NEG_ACT;
    v = (v - mean[c]) * rsqrtf(var[c] + BN_EPS) * g[c] + be[c];
    out[idx] = v;
}

// ---------------- final FC, only the last node's row is needed ------------
__global__ void fc_last_kernel(const float* __restrict__ h2, const float* __restrict__ Wfc,
                               const float* __restrict__ bfc, float* __restrict__ out,
                               int N, int K, int O) {
    int o = threadIdx.x;
    if (o >= O) return;
    const float* hr = h2 + (size_t)(N - 1) * K;
    float acc = bfc[o];
    for (int k = 0; k < K; ++k) acc += hr[k] * Wfc[(size_t)k * O + o];
    out[o] = acc;
}

// ---------------------------------------------------------------------------
static inline int nblk(long long n, int b) { return (int)((n + b - 1) / b); }

extern "C" void kernel_launch(void* const* d_in, const int* in_sizes, int n_in,
                              void* d_out, int out_size, void* d_ws, size_t ws_size,
                              hipStream_t stream) {
    const float* x       = (const float*)d_in[0];
    const int*   ei      = (const int*)  d_in[1];
    const float* W1      = (const float*)d_in[2];
    const float* a_src1  = (const float*)d_in[3];
    const float* a_dst1  = (const float*)d_in[4];
    const float* b1      = (const float*)d_in[5];
    const float* bn1_g   = (const float*)d_in[6];
    const float* bn1_b   = (const float*)d_in[7];
    const float* bn1_m   = (const float*)d_in[8];
    const float* bn1_v   = (const float*)d_in[9];
    const float* W2      = (const float*)d_in[10];
    const float* a_src2  = (const float*)d_in[11];
    const float* a_dst2  = (const float*)d_in[12];
    const float* b2      = (const float*)d_in[13];
    const float* bn2_g   = (const float*)d_in[14];
    const float* bn2_b   = (const float*)d_in[15];
    const float* bn2_m   = (const float*)d_in[16];
    const float* bn2_v   = (const float*)d_in[17];
    const float* fcW     = (const float*)d_in[18];
    const float* fcb     = (const float*)d_in[19];

    const int FEAT = 128, C1 = 64, C2 = 128, H = 2, OUT = 60;
    const int N = in_sizes[0] / FEAT;        // 100000
    const int E = in_sizes[1] / 2;           // 1600000
    const int* srcA = ei;
    const int* dstA = ei + E;
    const long long ET = (long long)E + N;   // edges incl. self loops

    // ---- workspace carve-up (≈174 MB) ----
    char* w = (char*)d_ws;
    auto alloc = [&](size_t bytes) -> void* {
        void* p = (void*)w;
        w += (bytes + 255) & ~(size_t)255;
        return p;
    };
    float*    hx1   = (float*)   alloc((size_t)N * C1 * 4);
    float*    out1  = (float*)   alloc((size_t)N * C1 * 4);
    float*    hx2   = (float*)   alloc((size_t)N * C2 * 4);
    float*    out2  = (float*)   alloc((size_t)N * C2 * 4);
    float*    alpha = (float*)   alloc((size_t)ET * H * 4);
    float*    as1   = (float*)   alloc((size_t)N * H * 4);
    float*    ad1   = (float*)   alloc((size_t)N * H * 4);
    unsigned* mk1   = (unsigned*)alloc((size_t)N * H * 4);
    float*    s1    = (float*)   alloc((size_t)N * H * 4);
    float*    as2   = (float*)   alloc((size_t)N * H * 4);
    float*    ad2   = (float*)   alloc((size_t)N * H * 4);
    unsigned* mk2   = (unsigned*)alloc((size_t)N * H * 4);
    float*    s2    = (float*)   alloc((size_t)N * H * 4);

    const int B = 256;
    const long long waves = (long long)(N / 16) * 32;   // threads for GEMM strips

    // =============== Layer 1 ===============
    gemm_wmma_strip<4><<<nblk(waves, B), B, 0, stream>>>(x, W1, hx1, N, FEAT);
    scores_kernel<<<nblk((long long)N * H, B), B, 0, stream>>>(hx1, a_src1, a_dst1, as1, ad1, N, H, 32);
    fill_u32<<<nblk((long long)N * H, B), B, 0, stream>>>(mk1, 0u, (long long)N * H);
    fill_f32<<<nblk((long long)N * H, B), B, 0, stream>>>(s1, 0.f, (long long)N * H);
    fill_f32<<<nblk((long long)N * C1, B), B, 0, stream>>>(out1, 0.f, (long long)N * C1);
    edge_max_kernel<<<nblk(ET * H, B), B, 0, stream>>>(srcA, dstA, E, N, as1, ad1, mk1, H);
    edge_sum_kernel<<<nblk(ET * H, B), B, 0, stream>>>(srcA, dstA, E, N, as1, ad1, mk1, alpha, s1, H);
    edge_agg_kernel<<<nblk(ET * (C1 >> 2), B), B, 0, stream>>>(srcA, dstA, E, N, hx1, alpha, s1, out1, C1, 32, H);
    post_kernel<<<nblk((long long)N * C1, B), B, 0, stream>>>(out1, b1, bn1_g, bn1_b, bn1_m, bn1_v, N, C1);

    // =============== Layer 2 ===============
    gemm_wmma_strip<8><<<nblk(waves, B), B, 0, stream>>>(out1, W2, hx2, N, C1);
    scores_kernel<<<nblk((long long)N * H, B), B, 0, stream>>>(hx2, a_src2, a_dst2, as2, ad2, N, H, 64);
    fill_u32<<<nblk((long long)N * H, B), B, 0, stream>>>(mk2, 0u, (long long)N * H);
    fill_f32<<<nblk((long long)N * H, B), B, 0, stream>>>(s2, 0.f, (long long)N * H);
    fill_f32<<<nblk((long long)N * C2, B), B, 0, stream>>>(out2, 0.f, (long long)N * C2);
    edge_max_kernel<<<nblk(ET * H, B), B, 0, stream>>>(srcA, dstA, E, N, as2, ad2, mk2, H);
    edge_sum_kernel<<<nblk(ET * H, B), B, 0, stream>>>(srcA, dstA, E, N, as2, ad2, mk2, alpha, s2, H);
    edge_agg_kernel<<<nblk(ET * (C2 >> 2), B), B, 0, stream>>>(srcA, dstA, E, N, hx2, alpha, s2, out2, C2, 64, H);
    post_kernel<<<nblk((long long)N * C2, B), B, 0, stream>>>(out2, b2, bn2_g, bn2_b, bn2_m, bn2_v, N, C2);

    // =============== FC on the last node only ===============
    fc_last_kernel<<<1, 64, 0, stream>>>(out2, fcW, fcb, (float*)d_out, N, C2, OUT);
}